// APPNP_Net_68341519613984
// MI455X (gfx1250) — compile-verified
//
#include <hip/hip_runtime.h>

#define F_IN 512
#define HID  128
#define NCLS 40
#define KPROP 10
#define ALPHA 0.1f

typedef __attribute__((ext_vector_type(2))) float v2f;
typedef __attribute__((ext_vector_type(8))) float v8f;

// ---------------------------------------------------------------------------
// Preprocessing kernels
// ---------------------------------------------------------------------------
__global__ void k_init_ones(float* __restrict__ deg, int n) {
    int i = blockIdx.x * blockDim.x + threadIdx.x;
    if (i < n) deg[i] = 1.0f;  // self-loop contribution
}

__global__ void k_count_deg(const int* __restrict__ dst, float* __restrict__ deg, int e) {
    int i = blockIdx.x * blockDim.x + threadIdx.x;
    if (i < e) atomicAdd(&deg[dst[i]], 1.0f);
}

__global__ void k_deg_transform(float* __restrict__ dinv, float* __restrict__ selfw, int n) {
    int i = blockIdx.x * blockDim.x + threadIdx.x;
    if (i < n) {
        float d = dinv[i];          // holds deg
        dinv[i]  = rsqrtf(d);       // deg^-0.5
        selfw[i] = 1.0f / d;        // self-loop weight
    }
}

__global__ void k_edge_norm(const int* __restrict__ src, const int* __restrict__ dst,
                            const float* __restrict__ dinv, float* __restrict__ norm, int e) {
    int i = blockIdx.x * blockDim.x + threadIdx.x;
    if (i < e) norm[i] = dinv[src[i]] * dinv[dst[i]];
}

// ---------------------------------------------------------------------------
// GEMM1: hidden = relu(x @ W1 + b1), fp32 WMMA 16x16x4
// Block: 256 threads = 8 waves; block covers 16 rows, wave w covers cols [16w,16w+16)
// x tile (16x512) staged in LDS with padded stride 516 (bank-conflict-free A reads)
// ---------------------------------------------------------------------------
__global__ __launch_bounds__(256) void k_gemm1_relu(const float* __restrict__ x,
                                                    const float* __restrict__ W1,
                                                    const float* __restrict__ b1,
                                                    float* __restrict__ hidden) {
    __shared__ float lx[16 * 516];
    const int tid = threadIdx.x;
    const int m0  = blockIdx.x * 16;

    // cooperative coalesced load of 16x512 fp32 tile -> LDS (float4)
    for (int j = tid; j < (16 * F_IN) / 4; j += 256) {
        int row = j >> 7;            // 128 float4 per row
        int c4  = j & 127;
        float4 v = ((const float4*)(x + (size_t)(m0 + row) * F_IN))[c4];
        ((float4*)(lx + row * 516))[c4] = v;
    }
    __syncthreads();

    const int lane = tid & 31;
    const int wave = tid >> 5;       // n-tile 0..7
    const int half = lane >> 4;      // K-pair selector
    const int mn   = lane & 15;      // row (A) / col (B)
    const int n0   = wave * 16;

    v8f acc = {};
    #pragma unroll 4
    for (int k0 = 0; k0 < F_IN; k0 += 4) {
        int ka = k0 + 2 * half;
        v2f a = *(const v2f*)(lx + mn * 516 + ka);          // A[m][ka], A[m][ka+1]
        v2f b;
        b.x = W1[(size_t)ka * HID + n0 + mn];               // B[ka][n]
        b.y = W1[(size_t)(ka + 1) * HID + n0 + mn];         // B[ka+1][n]
        acc = __builtin_amdgcn_wmma_f32_16x16x4_f32(false, a, false, b,
                                                    (short)0, acc, false, false);
    }

    float bias = b1[n0 + mn];
    #pragma unroll
    for (int v = 0; v < 8; ++v) {
        int row = m0 + v + 8 * half;
        float val = acc[v] + bias;
        hidden[(size_t)row * HID + n0 + mn] = val > 0.0f ? val : 0.0f;
    }
}

// ---------------------------------------------------------------------------
// GEMM2: z = hidden @ W2 + b2  (cols padded 40 -> 48; 3 waves of a 128-thread block)
// Also seeds h0 = z into d_out.
// ---------------------------------------------------------------------------
__global__ __launch_bounds__(128) void k_gemm2(const float* __restrict__ hidden,
                                               const float* __restrict__ W2,
                                               const float* __restrict__ b2,
                                               float* __restrict__ z,
                                               float* __restrict__ h0) {
    __shared__ float lh[16 * 132];   // hidden tile, padded stride
    __shared__ float lw[128 * 48];   // W2 zero-padded to 48 cols
    const int tid = threadIdx.x;
    const int m0  = blockIdx.x * 16;

    for (int j = tid; j < 128 * 48; j += 128) {
        int k = j / 48;
        int c = j - k * 48;
        lw[j] = (c < NCLS) ? W2[k * NCLS + c] : 0.0f;
    }
    for (int j = tid; j < (16 * HID) / 4; j += 128) {
        int row = j >> 5;            // 32 float4 per row
        int c4  = j & 31;
        float4 v = ((const float4*)(hidden + (size_t)(m0 + row) * HID))[c4];
        ((float4*)(lh + row * 132))[c4] = v;
    }
    __syncthreads();

    const int wave = tid >> 5;
    if (wave < 3) {                  // wave-uniform: EXEC all-ones inside
        const int lane = tid & 31;
        const int half = lane >> 4;
        const int mn   = lane & 15;
        const int n0   = wave * 16;

        v8f acc = {};
        #pragma unroll 4
        for (int k0 = 0; k0 < HID; k0 += 4) {
            int ka = k0 + 2 * half;
            v2f a = *(const v2f*)(lh + mn * 132 + ka);
            v2f b;
            b.x = lw[ka * 48 + n0 + mn];
            b.y = lw[(ka + 1) * 48 + n0 + mn];
            acc = __builtin_amdgcn_wmma_f32_16x16x4_f32(false, a, false, b,
                                                        (short)0, acc, false, false);
        }

        int col = n0 + mn;
        if (col < NCLS) {
            float bias = b2[col];
            #pragma unroll
            for (int v = 0; v < 8; ++v) {
                int row = m0 + v + 8 * half;
                float val = acc[v] + bias;
                z [(size_t)row * NCLS + col] = val;
                h0[(size_t)row * NCLS + col] = val;   // h_0 = z
            }
        }
    }
}

// ---------------------------------------------------------------------------
// Propagation step kernels
// ---------------------------------------------------------------------------
__global__ void k_prop_init(const float* __restrict__ h, const float* __restrict__ selfw,
                            float* __restrict__ agg, int total) {
    int t = blockIdx.x * blockDim.x + threadIdx.x;
    if (t < total) agg[t] = h[t] * selfw[t / NCLS];   // self-loop term
}

__global__ void k_prop_edges(const int* __restrict__ src, const int* __restrict__ dst,
                             const float* __restrict__ norm, const float* __restrict__ h,
                             float* __restrict__ agg, int total) {
    int t = blockIdx.x * blockDim.x + threadIdx.x;
    if (t >= total) return;
    int e = t / NCLS;
    int c = t - e * NCLS;
    int s = src[e];
    int d = dst[e];
    atomicAdd(&agg[(size_t)d * NCLS + c], h[(size_t)s * NCLS + c] * norm[e]);
}

__global__ void k_prop_blend(const float* __restrict__ agg, const float* __restrict__ z,
                             float* __restrict__ h, int total) {
    int t = blockIdx.x * blockDim.x + threadIdx.x;
    if (t < total) h[t] = (1.0f - ALPHA) * agg[t] + ALPHA * z[t];
}

// ---------------------------------------------------------------------------
// Host launcher
// ---------------------------------------------------------------------------
extern "C" void kernel_launch(void* const* d_in, const int* in_sizes, int n_in,
                              void* d_out, int out_size, void* d_ws, size_t ws_size,
                              hipStream_t stream) {
    const float* x   = (const float*)d_in[0];
    const int*   ei  = (const int*)  d_in[1];
    const float* W1  = (const float*)d_in[2];
    const float* b1  = (const float*)d_in[3];
    const float* W2  = (const float*)d_in[4];
    const float* b2  = (const float*)d_in[5];

    const int N = in_sizes[0] / F_IN;
    const int E = in_sizes[1] / 2;
    const int* src = ei;
    const int* dst = ei + E;

    float* h = (float*)d_out;                 // [N x 40], holds h_k in place

    // workspace carve-up (all offsets multiples of 16B for float4 paths)
    float* dinv   = (float*)d_ws;             // N   (holds deg, then deg^-0.5)
    float* selfw  = dinv + N;                 // N
    float* norm   = selfw + N;                // E
    float* hidden = norm + E;                 // N * 128
    float* z      = hidden + (size_t)N * HID; // N * 40
    float* agg    = z + (size_t)N * NCLS;     // N * 40

    const int TB = 256;
    const int nNC = N * NCLS;                 // 4M
    const int nEC = E * NCLS;                 // 128M (< 2^31)

    // --- normalization prep ---
    k_init_ones    <<<(N + TB - 1) / TB, TB, 0, stream>>>(dinv, N);
    k_count_deg    <<<(E + TB - 1) / TB, TB, 0, stream>>>(dst, dinv, E);
    k_deg_transform<<<(N + TB - 1) / TB, TB, 0, stream>>>(dinv, selfw, N);
    k_edge_norm    <<<(E + TB - 1) / TB, TB, 0, stream>>>(src, dst, dinv, norm, E);

    // --- MLP encoder (fp32 WMMA) ---
    k_gemm1_relu<<<N / 16, 256, 0, stream>>>(x, W1, b1, hidden);
    k_gemm2     <<<N / 16, 128, 0, stream>>>(hidden, W2, b2, z, h);

    // --- K propagation steps ---
    for (int k = 0; k < KPROP; ++k) {
        k_prop_init <<<(nNC + TB - 1) / TB, TB, 0, stream>>>(h, selfw, agg, nNC);
        k_prop_edges<<<(nEC + TB - 1) / TB, TB, 0, stream>>>(src, dst, norm, h, agg, nEC);
        k_prop_blend<<<(nNC + TB - 1) / TB, TB, 0, stream>>>(agg, z, h, nNC);
    }
}